// GATLayer_24120536334773
// MI455X (gfx1250) — compile-verified
//
#include <hip/hip_runtime.h>
#include <hip/hip_bf16.h>

typedef __attribute__((ext_vector_type(2))) float v2f;
typedef __attribute__((ext_vector_type(8))) float v8f;

// ---------------------------------------------------------------------------
// Kernel 1: fused dual GEMM with fp32 WMMA (16x16x4).
//   xl  = x @ W                       -> workspace
//   out = x @ Wr + br + bias          -> d_out (init for scatter-add)
// Block = 256 threads = 8 waves. Each block owns a 16-row strip of x; wave w
// owns output columns [16w, 16w+16). K-loop over 128 in steps of 4.
// ---------------------------------------------------------------------------
__global__ void gat_gemm_wmma(const float* __restrict__ x,
                              const float* __restrict__ W,
                              const float* __restrict__ Wr,
                              const float* __restrict__ bias,
                              const float* __restrict__ br,
                              float* __restrict__ xl,
                              float* __restrict__ out,
                              int N) {
    const int wave  = threadIdx.x >> 5;        // 0..7 -> column tile
    const int lane  = threadIdx.x & 31;
    const int row0  = blockIdx.x * 16;
    const int mlo   = lane & 15;               // M (A) / N (B,C) sub-index
    const int khalf = (lane >> 4) * 2;         // 0 or 2: K pair this half-wave owns
    const int n     = wave * 16 + mlo;         // global output column (0..127)

    int arow = row0 + mlo;                     // A-matrix row for this lane
    if (arow >= N) arow = N - 1;               // clamp (keeps EXEC all-ones)
    const float* __restrict__ xrow = x + (size_t)arow * 128;

    v8f cw = {};
    v8f cr = {};
    #pragma unroll
    for (int k0 = 0; k0 < 128; k0 += 4) {
        // A frag: 16x4 fp32. lanes 0-15: K = k0+{0,1}; lanes 16-31: K = k0+{2,3}
        v2f a;
        a.x = xrow[k0 + khalf];
        a.y = xrow[k0 + khalf + 1];
        // B frags: 4x16 fp32, row-major W[k][n]
        v2f bw, bq;
        bw.x = W [(k0 + khalf)     * 128 + n];
        bw.y = W [(k0 + khalf + 1) * 128 + n];
        bq.x = Wr[(k0 + khalf)     * 128 + n];
        bq.y = Wr[(k0 + khalf + 1) * 128 + n];
        cw = __builtin_amdgcn_wmma_f32_16x16x4_f32(false, a, false, bw, (short)0, cw, false, false);
        cr = __builtin_amdgcn_wmma_f32_16x16x4_f32(false, a, false, bq, (short)0, cr, false, false);
    }

    const float bsum = bias[n] + br[n];
    const int moff = (lane < 16) ? 0 : 8;      // C/D layout: VGPR v -> M = v (+8 for hi lanes)
    #pragma unroll
    for (int v = 0; v < 8; ++v) {
        const int mrow = row0 + moff + v;
        if (mrow < N) {
            xl [(size_t)mrow * 128 + n] = cw[v];
            out[(size_t)mrow * 128 + n] = cr[v] + bsum;
        }
    }
}

// ---------------------------------------------------------------------------
// Kernel 2: per-(node, head) attention logit halves + softmax-state init.
//   a_i[n,h] = sum_d xl[n,h,d] * att_i[h,d];  a_j likewise.
//   amax = -inf, denom = 0.
// ---------------------------------------------------------------------------
__global__ void gat_logits(const float* __restrict__ xl,
                           const float* __restrict__ att_i,
                           const float* __restrict__ att_j,
                           float* __restrict__ a_i,
                           float* __restrict__ a_j,
                           float* __restrict__ amax,
                           float* __restrict__ denom,
                           int N) {
    const int gid = blockIdx.x * blockDim.x + threadIdx.x;
    if (gid >= N * 4) return;
    const int node = gid >> 2;
    const int h    = gid & 3;
    const float* __restrict__ p  = xl + (size_t)node * 128 + h * 32;
    const float* __restrict__ ai = att_i + h * 32;
    const float* __restrict__ aj = att_j + h * 32;
    float si = 0.f, sj = 0.f;
    #pragma unroll
    for (int d = 0; d < 32; ++d) {
        const float v = p[d];
        si += v * ai[d];
        sj += v * aj[d];
    }
    a_i[gid]   = si;
    a_j[gid]   = sj;
    amax[gid]  = -__builtin_inff();
    denom[gid] = 0.f;
}

// Order-preserving float atomic max (sign-split int/uint trick).
__device__ __forceinline__ void atomicMaxF(float* addr, float val) {
    if (val >= 0.f) {
        atomicMax((int*)addr, __float_as_int(val));
    } else {
        atomicMin((unsigned int*)addr, __float_as_uint(val));
    }
}

__device__ __forceinline__ void edge_rc(const int* __restrict__ ei, int E, int e,
                                        int& r, int& c) {
    if (e < E) { r = ei[e]; c = ei[E + e]; }
    else       { r = c = e - E; }              // appended self-loops
}

// ---------------------------------------------------------------------------
// Kernel 3: segment max of leaky-relu logits, grouped by destination (row).
// ---------------------------------------------------------------------------
__global__ void gat_segmax(const int* __restrict__ ei, int E, int Etot,
                           const float* __restrict__ a_i,
                           const float* __restrict__ a_j,
                           float* __restrict__ amax) {
    const int gid = blockIdx.x * blockDim.x + threadIdx.x;
    if (gid >= Etot * 4) return;
    const int e = gid >> 2;
    const int h = gid & 3;
    int r, c; edge_rc(ei, E, e, r, c);
    float al = a_i[r * 4 + h] + a_j[c * 4 + h];
    al = (al > 0.f) ? al : 0.2f * al;          // leaky relu, slope 0.2
    atomicMaxF(&amax[r * 4 + h], al);
}

// ---------------------------------------------------------------------------
// Kernel 4: ex = exp(alpha - amax[row]); store per edge; segment-sum denom.
// ---------------------------------------------------------------------------
__global__ void gat_exp(const int* __restrict__ ei, int E, int Etot,
                        const float* __restrict__ a_i,
                        const float* __restrict__ a_j,
                        const float* __restrict__ amax,
                        float* __restrict__ denom,
                        float* __restrict__ exbuf) {
    const int gid = blockIdx.x * blockDim.x + threadIdx.x;
    if (gid >= Etot * 4) return;
    const int e = gid >> 2;
    const int h = gid & 3;
    int r, c; edge_rc(ei, E, e, r, c);
    float al = a_i[r * 4 + h] + a_j[c * 4 + h];
    al = (al > 0.f) ? al : 0.2f * al;
    const float ex = expf(al - amax[r * 4 + h]);
    exbuf[gid] = ex;
    atomicAdd(&denom[r * 4 + h], ex);
}

// ---------------------------------------------------------------------------
// Kernel 5: weighted message scatter. One wave per edge; lane l handles the
// float4 at offset 4l of the 128-float message (head h = l>>3). xl and out
// both live in L2 (25.6 MB each), so gathers + f32 atomics stay on-chip.
// ---------------------------------------------------------------------------
__global__ void gat_scatter(const int* __restrict__ ei, int E, int Etot,
                            const float* __restrict__ xl,
                            const float* __restrict__ exbuf,
                            const float* __restrict__ denom,
                            float* __restrict__ out) {
    const int e = blockIdx.x * 8 + (threadIdx.x >> 5);
    if (e >= Etot) return;
    const int lane = threadIdx.x & 31;
    int r, c; edge_rc(ei, E, e, r, c);
    const int h = lane >> 3;
    const float coeff = exbuf[e * 4 + h] / (denom[r * 4 + h] + 1e-16f);
    const float4 v = *((const float4*)(xl + (size_t)c * 128) + lane);
    float* dst = out + (size_t)r * 128 + lane * 4;
    atomicAdd(dst + 0, v.x * coeff);
    atomicAdd(dst + 1, v.y * coeff);
    atomicAdd(dst + 2, v.z * coeff);
    atomicAdd(dst + 3, v.w * coeff);
}

// ---------------------------------------------------------------------------
extern "C" void kernel_launch(void* const* d_in, const int* in_sizes, int n_in,
                              void* d_out, int out_size, void* d_ws, size_t ws_size,
                              hipStream_t stream) {
    const float* x     = (const float*)d_in[0];
    const int*   ei    = (const int*)  d_in[1];   // [2, E]: row then col
    const float* W     = (const float*)d_in[2];
    const float* att_i = (const float*)d_in[3];
    const float* att_j = (const float*)d_in[4];
    const float* bias  = (const float*)d_in[5];
    const float* Wr    = (const float*)d_in[6];
    const float* br    = (const float*)d_in[7];
    float*       out   = (float*)d_out;

    const int N    = in_sizes[0] / 128;
    const int E    = in_sizes[1] / 2;
    const int Etot = E + N;                        // edges + self-loops

    // workspace carve-out (~42.5 MB total)
    char*  ws    = (char*)d_ws;
    float* xl    = (float*)ws; ws += (size_t)N * 128 * sizeof(float);
    float* a_i   = (float*)ws; ws += (size_t)N * 4 * sizeof(float);
    float* a_j   = (float*)ws; ws += (size_t)N * 4 * sizeof(float);
    float* amax  = (float*)ws; ws += (size_t)N * 4 * sizeof(float);
    float* denom = (float*)ws; ws += (size_t)N * 4 * sizeof(float);
    float* exbuf = (float*)ws; // (E+N)*4 floats

    const int rowTiles = (N + 15) / 16;
    gat_gemm_wmma<<<rowTiles, 256, 0, stream>>>(x, W, Wr, bias, br, xl, out, N);

    gat_logits<<<(N * 4 + 255) / 256, 256, 0, stream>>>(xl, att_i, att_j,
                                                        a_i, a_j, amax, denom, N);

    const int egrid = (Etot * 4 + 255) / 256;
    gat_segmax<<<egrid, 256, 0, stream>>>(ei, E, Etot, a_i, a_j, amax);
    gat_exp   <<<egrid, 256, 0, stream>>>(ei, E, Etot, a_i, a_j, amax, denom, exbuf);

    gat_scatter<<<(Etot + 7) / 8, 256, 0, stream>>>(ei, E, Etot, xl, exbuf, denom, out);
}